// LatentRegulaizer_63488206570034
// MI455X (gfx1250) — compile-verified
//
#include <hip/hip_runtime.h>
#include <hip/hip_bf16.h>

typedef float v2f __attribute__((ext_vector_type(2)));
typedef float v8f __attribute__((ext_vector_type(8)));

#define PATCH_SZ 14
#define Bc 2
#define Tc 8
#define Mc 900
#define Hc 27
#define Pc 729          // H*W patches
#define Dc 1024
#define NTRACK (Bc * Tc * Mc)          // 14400
#define NUNIT  (Bc * (Tc - 1) * Mc)    // 12600
#define NTILE  ((Pc + 15) / 16)        // 46

// ---------------------------------------------------------------------------
// Kernel 1: nearest patch center per track via V_WMMA_F32_16X16X4_F32.
// One wave (32 lanes) handles 16 flattened tracks. A = tracks[16 x 4] (K=2
// real + 2 zero pad), B = patch centers [4 x 16] per tile, D = dot products.
// d2 = (||t||^2 - 2*dot) + ||p||^2, running argmin over 46 tiles, then a
// 16-lane shfl-xor argmin (lowest-index tie break, matching jnp.argmin).
// Min updates are branchless (v_cndmask) to keep the tile loop tight.
// ---------------------------------------------------------------------------
__global__ __launch_bounds__(32)
void lr_nn_idx_kernel(const float* __restrict__ tracks, int* __restrict__ nn_idx)
{
    const int lane = threadIdx.x;          // 0..31
    const int base = blockIdx.x * 16;      // first flat track of this group
    const int half = lane >> 4;            // 0: rows 0..7, 1: rows 8..15
    const int n    = lane & 15;            // column within 16-wide patch tile

    // A matrix: lanes 0-15 hold (x,y) of track base+lane for K=0,1.
    // Lanes 16-31 hold K=2,3 = zero padding.
    v2f A = {0.0f, 0.0f};
    float my_t2 = 0.0f;
    if (lane < 16) {
        float tx = tracks[(size_t)(base + lane) * 2 + 0];
        float ty = tracks[(size_t)(base + lane) * 2 + 1];
        A.x = tx; A.y = ty;
        my_t2 = tx * tx + ty * ty;
    }

    __shared__ float s_t2[16];
    if (lane < 16) s_t2[lane] = my_t2;
    __syncthreads();

    float t2row[8];
#pragma unroll
    for (int v = 0; v < 8; ++v) t2row[v] = s_t2[v + 8 * half];

    float best_d2[8];
    int   best_idx[8];
#pragma unroll
    for (int v = 0; v < 8; ++v) { best_d2[v] = 3.402823466e38f; best_idx[v] = 0x7fffffff; }

    for (int tile = 0; tile < NTILE; ++tile) {
        const int p = tile * 16 + n;                 // this lane's patch column
        const int prow = p / Hc, pcol = p % Hc;
        const float px = (float)pcol * (float)PATCH_SZ + (float)(PATCH_SZ / 2);
        const float py = (float)prow * (float)PATCH_SZ + (float)(PATCH_SZ / 2);

        // B matrix: lanes 0-15 carry rows K=0 (px) and K=1 (py) of column n;
        // lanes 16-31 carry zero-padded rows K=2,3.
        v2f Bm = {0.0f, 0.0f};
        if (half == 0) { Bm.x = px; Bm.y = py; }

        v8f cz = {0.f, 0.f, 0.f, 0.f, 0.f, 0.f, 0.f, 0.f};
        v8f dot = __builtin_amdgcn_wmma_f32_16x16x4_f32(
            /*neg_a=*/false, A, /*neg_b=*/false, Bm,
            /*c_mod=*/(short)0, cz, /*reuse_a=*/false, /*reuse_b=*/false);

        const float p2 = px * px + py * py;
        const bool valid = (p < Pc);
#pragma unroll
        for (int v = 0; v < 8; ++v) {
            // D layout: VGPR v, this lane -> (M = v + 8*half, N = n)
            const float d2 = (t2row[v] - 2.0f * dot[v]) + p2;
            const bool take = valid & ((d2 < best_d2[v]) |
                                       ((d2 == best_d2[v]) & (p < best_idx[v])));
            best_d2[v]  = take ? d2 : best_d2[v];
            best_idx[v] = take ? p  : best_idx[v];
        }
    }

    // Argmin across the 16 lanes of each half (columns), per row.
#pragma unroll
    for (int v = 0; v < 8; ++v) {
        float d = best_d2[v];
        int   idx = best_idx[v];
#pragma unroll
        for (int off = 8; off >= 1; off >>= 1) {
            const float od = __shfl_xor(d, off, 16);
            const int   oi = __shfl_xor(idx, off, 16);
            const bool take = (od < d) | ((od == d) & (oi < idx));
            d   = take ? od : d;
            idx = take ? oi : idx;
        }
        if (n == 0) nn_idx[base + v + 8 * half] = idx;
    }
}

// ---------------------------------------------------------------------------
// Kernel 2: fused masked-L1 gather. One 256-thread block per (b, t in [1,T),
// m). Thread tid covers float4 at element offset tid*4 of D=1024. pf[t] is
// loaded once and reused for both the consecutive and reference terms.
// Masks are uniform per block, so the skip branches are wave-coherent.
// Writes one partial sum per block (deterministic reduction, no atomics).
// ---------------------------------------------------------------------------
__global__ __launch_bounds__(256)
void lr_l1_kernel(const float* __restrict__ features,
                  const int*   __restrict__ nn_idx,
                  const int*   __restrict__ vis,
                  float* __restrict__ partial)
{
    const int u   = blockIdx.x;            // 0 .. NUNIT-1
    const int tid = threadIdx.x;
    const int m  = u % Mc;
    const int r  = u / Mc;
    const int t1 = r % (Tc - 1) + 1;       // frame t in [1, T)
    const int b  = r / (Tc - 1);

    const int i_t  = (b * Tc + t1)     * Mc + m;
    const int i_tm = (b * Tc + t1 - 1) * Mc + m;
    const int i_0  = (b * Tc + 0)      * Mc + m;

    const bool v_t = vis[i_t] != 0;
    const bool mc  = v_t && (vis[i_tm] != 0);
    const bool mr  = v_t && (vis[i_0]  != 0);

    float val = 0.0f;
    if (mc || mr) {
        const float4* rt  = (const float4*)(features +
            ((size_t)(b * Tc + t1) * Pc + nn_idx[i_t]) * Dc);
        float4 a = rt[tid];
        if (mc) {
            const float4* rtm = (const float4*)(features +
                ((size_t)(b * Tc + t1 - 1) * Pc + nn_idx[i_tm]) * Dc);
            float4 x = rtm[tid];
            val += fabsf(a.x - x.x) + fabsf(a.y - x.y)
                 + fabsf(a.z - x.z) + fabsf(a.w - x.w);
        }
        if (mr) {
            const float4* r0 = (const float4*)(features +
                ((size_t)(b * Tc + 0) * Pc + nn_idx[i_0]) * Dc);
            float4 x = r0[tid];
            val += fabsf(a.x - x.x) + fabsf(a.y - x.y)
                 + fabsf(a.z - x.z) + fabsf(a.w - x.w);
        }
    }

    __shared__ float red[256];
    red[tid] = val;
    __syncthreads();
#pragma unroll
    for (int s = 128; s > 0; s >>= 1) {
        if (tid < s) red[tid] += red[tid + s];
        __syncthreads();
    }
    if (tid == 0) partial[u] = red[0];
}

// ---------------------------------------------------------------------------
// Kernel 3: deterministic final reduction of NUNIT partials, scale by
// 0.01 * (1/NUNIT) (both means share the same denominator).
// ---------------------------------------------------------------------------
__global__ __launch_bounds__(256)
void lr_finalize_kernel(const float* __restrict__ partial, float* __restrict__ out)
{
    const int tid = threadIdx.x;
    float s = 0.0f;
    for (int i = tid; i < NUNIT; i += 256) s += partial[i];
    __shared__ float red[256];
    red[tid] = s;
    __syncthreads();
#pragma unroll
    for (int k = 128; k > 0; k >>= 1) {
        if (tid < k) red[tid] += red[tid + k];
        __syncthreads();
    }
    if (tid == 0) out[0] = red[0] * (0.01f / (float)NUNIT);
}

extern "C" void kernel_launch(void* const* d_in, const int* in_sizes, int n_in,
                              void* d_out, int out_size, void* d_ws, size_t ws_size,
                              hipStream_t stream)
{
    (void)in_sizes; (void)n_in; (void)out_size; (void)ws_size;

    const float* features = (const float*)d_in[0];   // [B*T, 729, 1024] f32
    const float* tracks   = (const float*)d_in[1];   // [B, T, M, 2]     f32
    const int*   vis      = (const int*)d_in[2];     // [B, T, M]        int
    float* out = (float*)d_out;                      // scalar f32

    int*   nn      = (int*)d_ws;                                   // 14400 ints
    float* partial = (float*)((char*)d_ws + NTRACK * sizeof(int)); // 12600 floats

    lr_nn_idx_kernel<<<NTRACK / 16, 32, 0, stream>>>(tracks, nn);
    lr_l1_kernel<<<NUNIT, 256, 0, stream>>>(features, nn, vis, partial);
    lr_finalize_kernel<<<1, 256, 0, stream>>>(partial, out);
}